// SafeRocketLeagueGCN_4131758539251
// MI455X (gfx1250) — compile-verified
//
#include <hip/hip_runtime.h>
#include <math.h>

#define NG   262144            // graphs
#define NPG  6                 // nodes per graph
#define NND  (NG * NPG)        // 1572864 nodes
#define FDIM 13
#define HDIM 32
#define GDIM 9
#define EPS  1e-5f

#define WAVES 4                // waves per block
#define GPW   8                // graphs per wave
#define NW    48               // nodes per wave (GPW*NPG)
#define GPB   (WAVES * GPW)    // graphs per block = 32
#define CONV_BLOCKS (NG / GPB) // 8192

typedef __attribute__((ext_vector_type(2))) float v2f;
typedef __attribute__((ext_vector_type(8))) float v8f;

// position of target j inside source-row s's 5 outgoing edges (meshgrid order, s!=t)
__device__ __forceinline__ int eoff(int s, int j) { return s * 5 + (j - (j > s ? 1 : 0)); }

__global__ void k_zero(float* __restrict__ p, int n) {
    int i = blockIdx.x * blockDim.x + threadIdx.x;
    if (i < n) p[i] = 0.f;
}

// ---------------- conv1: h = Agg( x @ W1 ) + b1 ; accumulate BN stats ----------------
__global__ __launch_bounds__(128) void k_conv1(const float* __restrict__ x,
                                               const float* __restrict__ ew,
                                               const float* __restrict__ W1,
                                               const float* __restrict__ b1,
                                               float* __restrict__ h,
                                               float* __restrict__ stats) {
    __shared__ float Wl[16 * 32];              // W1 padded K 13 -> 16
    __shared__ float xs[WAVES][NW * 16];       // input tiles (K padded)
    __shared__ float ts[WAVES][NW * 32];       // linear output tiles
    __shared__ float ews[WAVES][GPW * 30];
    __shared__ float dvs[WAVES][GPW * 6];
    __shared__ float red[WAVES * 64];

    const int tid  = threadIdx.x;
    const int wave = tid >> 5;
    const int lane = tid & 31;
    const int half = lane >> 4;
    const int lm   = lane & 15;

    const int g0 = blockIdx.x * GPB + wave * GPW;   // first graph of this wave
    const int n0 = g0 * NPG;                        // first node of this wave

    // stage W1 (block-wide), zero-padded rows 13..15
    for (int it = 0; it < 4; ++it) {
        int idx = tid + it * 128;                   // 512 total
        int k = idx >> 5, c = idx & 31;
        Wl[idx] = (k < FDIM) ? W1[k * 32 + c] : 0.f;
    }
    // stage x tile, zero-padded cols 13..15
    for (int it = 0; it < 24; ++it) {
        int idx = lane + it * 32;                   // 768 total
        int r = idx >> 4, k = idx & 15;
        xs[wave][idx] = (k < FDIM) ? x[(n0 + r) * FDIM + k] : 0.f;
    }
    // stage edge weights (30 per graph)
    for (int it = 0; it < 8; ++it) {
        int idx = lane + it * 32;
        if (idx < GPW * 30) ews[wave][idx] = ew[g0 * 30 + idx];
    }
    __syncthreads();

    // degrees (self-loop weight 1) -> dinv
    for (int it = 0; it < 2; ++it) {
        int idx = lane + it * 32;
        if (idx < GPW * 6) {
            int g = idx / 6, j = idx % 6;
            float deg = 1.f;
            #pragma unroll
            for (int s = 0; s < 6; ++s)
                if (s != j) deg += ews[wave][g * 30 + eoff(s, j)];
            dvs[wave][idx] = rsqrtf(deg);
        }
    }

    // WMMA: ts[48x32] = xs[48x16] @ Wl[16x32]  (f32 16x16x4, 3 M-tiles x 2 N-tiles x 4 K-chunks)
    for (int mt = 0; mt < 3; ++mt) {
        const int mbase = mt * 16;
        #pragma unroll
        for (int nt = 0; nt < 2; ++nt) {
            v8f acc = {};
            #pragma unroll
            for (int k0 = 0; k0 < 16; k0 += 4) {
                const int kb = k0 + (half << 1);
                v2f a, b;
                a.x = xs[wave][(mbase + lm) * 16 + kb];
                a.y = xs[wave][(mbase + lm) * 16 + kb + 1];
                b.x = Wl[kb * 32 + nt * 16 + lm];
                b.y = Wl[(kb + 1) * 32 + nt * 16 + lm];
                acc = __builtin_amdgcn_wmma_f32_16x16x4_f32(false, a, false, b,
                                                            (short)0, acc, false, false);
            }
            #pragma unroll
            for (int r = 0; r < 8; ++r)
                ts[wave][(mbase + r + half * 8) * 32 + nt * 16 + lm] = acc[r];
        }
    }
    __syncthreads();

    // per-graph 6x6 normalized-adjacency aggregation; lane == output channel
    const int c = lane;
    const float bc = b1[c];
    float lsum = 0.f, lsq = 0.f;
    for (int gg = 0; gg < GPW; ++gg) {
        const float* tg = &ts[wave][gg * 6 * 32];
        const float* dv = &dvs[wave][gg * 6];
        const float* eg = &ews[wave][gg * 30];
        #pragma unroll
        for (int j = 0; j < 6; ++j) {
            float dj = dv[j];
            float acc = dj * dj * tg[j * 32 + c];
            #pragma unroll
            for (int i = 0; i < 6; ++i)
                if (i != j) acc += dv[i] * eg[eoff(i, j)] * dj * tg[i * 32 + c];
            float v = acc + bc;
            h[(n0 + gg * 6 + j) * 32 + c] = v;
            lsum += v; lsq += v * v;
        }
    }
    red[wave * 64 + c]      = lsum;
    red[wave * 64 + 32 + c] = lsq;
    __syncthreads();
    if (tid < 64) {
        float s = 0.f;
        #pragma unroll
        for (int w = 0; w < WAVES; ++w) s += red[w * 64 + tid];
        atomicAdd(&stats[tid], s);   // [0..31]=sum, [32..63]=sumsq
    }
}

// ------------- finalize BN params: scale = g*rsqrt(var+eps); shift = be - mean*scale -------------
__global__ void k_bnfin(const float* __restrict__ stats, const float* __restrict__ g,
                        const float* __restrict__ be, float* __restrict__ bnp) {
    int c = threadIdx.x;
    if (c < 32) {
        const float invN = 1.f / (float)NND;
        float mean  = stats[c] * invN;
        float var   = stats[32 + c] * invN - mean * mean;
        float scale = g[c] * rsqrtf(var + EPS);
        bnp[c]      = scale;
        bnp[32 + c] = be[c] - mean * scale;
    }
}

// ------- conv2 (in place): h = Agg( relu(bn1(h)) @ W2 ) + b2 ; accumulate BN2 stats -------
__global__ __launch_bounds__(128) void k_conv2(const float* __restrict__ ew,
                                               const float* __restrict__ W2,
                                               const float* __restrict__ b2,
                                               const float* __restrict__ bnp,
                                               float* __restrict__ h,
                                               float* __restrict__ stats) {
    __shared__ float Wl[32 * 32];
    __shared__ float as[WAVES][NW * 32];
    __shared__ float ts[WAVES][NW * 32];
    __shared__ float ews[WAVES][GPW * 30];
    __shared__ float dvs[WAVES][GPW * 6];
    __shared__ float red[WAVES * 64];

    const int tid  = threadIdx.x;
    const int wave = tid >> 5;
    const int lane = tid & 31;
    const int half = lane >> 4;
    const int lm   = lane & 15;

    const int g0 = blockIdx.x * GPB + wave * GPW;
    const int n0 = g0 * NPG;

    for (int it = 0; it < 8; ++it) {            // 1024 elems of W2
        int idx = tid + it * 128;
        Wl[idx] = W2[idx];
    }
    // stage relu(bn1(h)) tile
    for (int it = 0; it < 48; ++it) {
        int idx = lane + it * 32;               // 1536 total
        int c2 = idx & 31;
        float v = h[n0 * 32 + idx];
        v = v * bnp[c2] + bnp[32 + c2];
        as[wave][idx] = fmaxf(v, 0.f);
    }
    for (int it = 0; it < 8; ++it) {
        int idx = lane + it * 32;
        if (idx < GPW * 30) ews[wave][idx] = ew[g0 * 30 + idx];
    }
    __syncthreads();

    for (int it = 0; it < 2; ++it) {
        int idx = lane + it * 32;
        if (idx < GPW * 6) {
            int g = idx / 6, j = idx % 6;
            float deg = 1.f;
            #pragma unroll
            for (int s = 0; s < 6; ++s)
                if (s != j) deg += ews[wave][g * 30 + eoff(s, j)];
            dvs[wave][idx] = rsqrtf(deg);
        }
    }

    // WMMA: ts[48x32] = as[48x32] @ W2[32x32]  (8 K-chunks)
    for (int mt = 0; mt < 3; ++mt) {
        const int mbase = mt * 16;
        #pragma unroll
        for (int nt = 0; nt < 2; ++nt) {
            v8f acc = {};
            #pragma unroll
            for (int k0 = 0; k0 < 32; k0 += 4) {
                const int kb = k0 + (half << 1);
                v2f a, b;
                a.x = as[wave][(mbase + lm) * 32 + kb];
                a.y = as[wave][(mbase + lm) * 32 + kb + 1];
                b.x = Wl[kb * 32 + nt * 16 + lm];
                b.y = Wl[(kb + 1) * 32 + nt * 16 + lm];
                acc = __builtin_amdgcn_wmma_f32_16x16x4_f32(false, a, false, b,
                                                            (short)0, acc, false, false);
            }
            #pragma unroll
            for (int r = 0; r < 8; ++r)
                ts[wave][(mbase + r + half * 8) * 32 + nt * 16 + lm] = acc[r];
        }
    }
    __syncthreads();

    const int c = lane;
    const float bc = b2[c];
    float lsum = 0.f, lsq = 0.f;
    for (int gg = 0; gg < GPW; ++gg) {
        const float* tg = &ts[wave][gg * 6 * 32];
        const float* dv = &dvs[wave][gg * 6];
        const float* eg = &ews[wave][gg * 30];
        #pragma unroll
        for (int j = 0; j < 6; ++j) {
            float dj = dv[j];
            float acc = dj * dj * tg[j * 32 + c];
            #pragma unroll
            for (int i = 0; i < 6; ++i)
                if (i != j) acc += dv[i] * eg[eoff(i, j)] * dj * tg[i * 32 + c];
            float v = acc + bc;
            h[(n0 + gg * 6 + j) * 32 + c] = v;   // in-place overwrite (own rows only)
            lsum += v; lsq += v * v;
        }
    }
    red[wave * 64 + c]      = lsum;
    red[wave * 64 + 32 + c] = lsq;
    __syncthreads();
    if (tid < 64) {
        float s = 0.f;
        #pragma unroll
        for (int w = 0; w < WAVES; ++w) s += red[w * 64 + tid];
        atomicAdd(&stats[tid], s);
    }
}

// ---------------- pool + heads: mean over 6 nodes of relu(bn2(h)), two sigmoid dots ----------------
__global__ __launch_bounds__(256) void k_head(const float* __restrict__ h,
                                              const float* __restrict__ gf,
                                              const float* __restrict__ bnp2,
                                              const float* __restrict__ Wo, const float* __restrict__ bo,
                                              const float* __restrict__ Wb, const float* __restrict__ bb,
                                              float* __restrict__ out) {
    const int tid  = threadIdx.x;
    const int wave = tid >> 5;
    const int lane = tid & 31;
    const int g0   = blockIdx.x * 64 + wave * 8;

    const float sc = bnp2[lane], sh = bnp2[32 + lane];
    const float wo = Wo[lane],   wb = Wb[lane];

    for (int gg = 0; gg < 8; ++gg) {
        const int g = g0 + gg;
        const float* hg = &h[g * NPG * 32];
        float s = 0.f;
        #pragma unroll
        for (int j = 0; j < NPG; ++j) {
            float v = hg[j * 32 + lane] * sc + sh;
            s += fmaxf(v, 0.f);
        }
        s *= (1.f / 6.f);
        float po = s * wo, pb = s * wb;
        #pragma unroll
        for (int off = 16; off > 0; off >>= 1) {
            po += __shfl_xor(po, off, 32);
            pb += __shfl_xor(pb, off, 32);
        }
        if (lane == 0) {
            float zo = po + bo[0], zb = pb + bb[0];
            #pragma unroll
            for (int k = 0; k < GDIM; ++k) {
                float gv = gf[g * GDIM + k];
                zo += gv * Wo[32 + k];
                zb += gv * Wb[32 + k];
            }
            out[g]      = 1.f / (1.f + __expf(-zo));
            out[NG + g] = 1.f / (1.f + __expf(-zb));
        }
    }
}

extern "C" void kernel_launch(void* const* d_in, const int* in_sizes, int n_in,
                              void* d_out, int out_size, void* d_ws, size_t ws_size,
                              hipStream_t stream) {
    const float* x   = (const float*)d_in[0];
    const float* ew  = (const float*)d_in[2];
    const float* gf  = (const float*)d_in[4];
    const float* W1  = (const float*)d_in[5];
    const float* b1  = (const float*)d_in[6];
    const float* g1  = (const float*)d_in[7];
    const float* be1 = (const float*)d_in[8];
    const float* W2  = (const float*)d_in[9];
    const float* b2  = (const float*)d_in[10];
    const float* g2  = (const float*)d_in[11];
    const float* be2 = (const float*)d_in[12];
    const float* Wo  = (const float*)d_in[13];
    const float* bo  = (const float*)d_in[14];
    const float* Wb  = (const float*)d_in[15];
    const float* bb  = (const float*)d_in[16];
    float* out = (float*)d_out;

    float* ws     = (float*)d_ws;
    float* h      = ws;                              // [NND*32] pre-BN activations (reused in place)
    float* stats1 = ws + (size_t)NND * HDIM;         // 64
    float* stats2 = stats1 + 64;                     // 64
    float* bnp1   = stats1 + 128;                    // 64
    float* bnp2   = stats1 + 192;                    // 64

    k_zero <<<1, 256, 0, stream>>>(stats1, 256);
    k_conv1<<<CONV_BLOCKS, 128, 0, stream>>>(x, ew, W1, b1, h, stats1);
    k_bnfin<<<1, 32, 0, stream>>>(stats1, g1, be1, bnp1);
    k_conv2<<<CONV_BLOCKS, 128, 0, stream>>>(ew, W2, b2, bnp1, h, stats2);
    k_bnfin<<<1, 32, 0, stream>>>(stats2, g2, be2, bnp2);
    k_head <<<NG / 64, 256, 0, stream>>>(h, gf, bnp2, Wo, bo, Wb, bb, out);
}